// Model_4432406250106
// MI455X (gfx1250) — compile-verified
//
#include <hip/hip_runtime.h>

#define N_OSC     2048
#define N_SAMPLES 16384

typedef __attribute__((ext_vector_type(2))) float v2f;
typedef __attribute__((ext_vector_type(8))) float v8f;

// ---------------------------------------------------------------------------
// Prep: amp[o] = ||params[o]||_2 ; fh[o] = atan2(p0, amp) / (2*pi)
// phase(i) = (i+1)*atan2(p0,amp)  ->  v_sin input (revolutions) = (i+1)*fh[o]
// ---------------------------------------------------------------------------
__global__ __launch_bounds__(256) void osc_prep(const float* __restrict__ params,
                                                float* __restrict__ fh,
                                                float* __restrict__ am) {
    int o = blockIdx.x * blockDim.x + threadIdx.x;
    if (o < N_OSC) {
        float p0  = params[2 * o + 0];
        float p1  = params[2 * o + 1];
        float amp = sqrtf(p0 * p0 + p1 * p1);
        fh[o] = atan2f(p0, amp) * 0.15915494309189535f;  // 1/(2*pi)
        am[o] = amp;
    }
}

// ---------------------------------------------------------------------------
// 4-oscillator x 16-sample step, reduced across oscillators with
// V_WMMA_F32_16X16X4_F32 (A = ones => D[m][n] = sum_k B[k][n] + C[m][n]).
// B layout (4x16 f32, 2 VGPRs): v0 = K0 (lanes 0-15) / K2 (lanes 16-31),
//                               v1 = K1 (lanes 0-15) / K3 (lanes 16-31).
// ---------------------------------------------------------------------------
__device__ __forceinline__ v8f step4(int c,
                                     const float* __restrict__ fh,
                                     const float* __restrict__ am,
                                     float m, bool hi, v2f ones, v8f acc) {
    // wave-uniform scalar loads of the 4 oscillators' params
    float f0 = fh[c + 0], f1 = fh[c + 1], f2 = fh[c + 2], f3 = fh[c + 3];
    float a0 = am[c + 0], a1 = am[c + 1], a2 = am[c + 2], a3 = am[c + 3];
    float fA = hi ? f2 : f0;
    float fB = hi ? f3 : f1;
    float aA = hi ? a2 : a0;
    float aB = hi ? a3 : a1;
    // sin(2*pi*u) with u in revolutions: fract then hw sin
    float u0 = __builtin_amdgcn_fractf(m * fA);
    float u1 = __builtin_amdgcn_fractf(m * fB);
    v2f b;
    b.x = __builtin_amdgcn_sinf(u0) * aA;
    b.y = __builtin_amdgcn_sinf(u1) * aB;
    return __builtin_amdgcn_wmma_f32_16x16x4_f32(false, ones, false, b,
                                                 (short)0, acc, false, false);
}

__global__ __launch_bounds__(256) void osc_synth(const float* __restrict__ fh,
                                                 const float* __restrict__ am,
                                                 float* __restrict__ out) {
    const int  lane = threadIdx.x & 31;
    const int  wave = threadIdx.x >> 5;
    const int  base = (blockIdx.x * 8 + wave) * 16;  // 16 samples per wave
    const int  n    = lane & 15;
    const bool hi   = lane >= 16;
    const float m   = (float)(base + n + 1);          // (i+1) factor

    v2f ones; ones.x = 1.0f; ones.y = 1.0f;
    v8f acc0 = {};
    v8f acc1 = {};

    // 2048 oscillators, 8 per iteration, two accumulators to hide WMMA latency
    for (int c = 0; c < N_OSC; c += 8) {
        acc0 = step4(c + 0, fh, am, m, hi, ones, acc0);
        acc1 = step4(c + 4, fh, am, m, hi, ones, acc1);
    }

    // D[m][n] is replicated across rows m; lanes 0-15 hold N=0..15 in VGPR0
    float r = acc0[0] + acc1[0];
    if (lane < 16) out[base + n] = r;
}

extern "C" void kernel_launch(void* const* d_in, const int* in_sizes, int n_in,
                              void* d_out, int out_size, void* d_ws, size_t ws_size,
                              hipStream_t stream) {
    // d_in[0] = x (1 x 16384 f32, unused by the math); d_in[1] = params (2048 x 2 f32)
    const float* params = (const float*)d_in[1];
    float* fh  = (float*)d_ws;       // 2048 floats: theta/(2*pi)
    float* am  = fh + N_OSC;         // 2048 floats: amplitude
    float* out = (float*)d_out;      // 16384 floats

    osc_prep<<<(N_OSC + 255) / 256, 256, 0, stream>>>(params, fh, am);
    osc_synth<<<N_SAMPLES / 128, 256, 0, stream>>>(fh, am, out);
}